// GNFConverter_61770219651343
// MI455X (gfx1250) — compile-verified
//
#include <hip/hip_runtime.h>
#include <math.h>

// CDNA5 / gfx1250: wave32, WMMA 16x16x4 F32.
typedef float v2f __attribute__((ext_vector_type(2)));
typedef float v8f __attribute__((ext_vector_type(8)));

#define MAX_ATOMS 128
#define WAVES_PER_BLOCK 8
#define L2E 1.44269504088896340736f   // log2(e)

// One 16-atom group. The WMMA computes the COMPLETE exp2 argument:
//   A = [L2E*qx, L2E*qy, L2E*qz, 1]  (16x4)
//   B = [cx, cy, cz, hc2]^T          (4x16), hc2 = -L2E/2*|c|^2
//   C = hq2 (loop-invariant),        hq2 = -L2E/2*|q|^2
//   S[m][n] = L2E*(q.c) - L2E/2*|c|^2 - L2E/2*|q|^2  ->  w = exp2(S)
__device__ __forceinline__ void gnf_group(const float4& c4, const v2f& A2, int hi,
                                          const v8f& hq2v,
                                          float ax[8], float ay[8], float az[8], float aw[8])
{
    // B operand (4x16 f32): VGPR0 = K0|K2 rows, VGPR1 = K1|K3 rows.
    v2f B;
    B.x = hi ? c4.z : c4.x;
    B.y = hi ? c4.w : c4.y;

    v8f S = __builtin_amdgcn_wmma_f32_16x16x4_f32(false, A2, false, B, (short)0, hq2v,
                                                  false, false);
#pragma unroll
    for (int r = 0; r < 8; ++r) {
        float w = __builtin_amdgcn_exp2f(S[r]);   // raw v_exp_f32
        ax[r] = fmaf(w, c4.x, ax[r]);
        ay[r] = fmaf(w, c4.y, ay[r]);
        az[r] = fmaf(w, c4.z, az[r]);
        aw[r] += w;
    }
}

__global__ __launch_bounds__(256)
void gnf_field_kernel(const float* __restrict__ coords,
                      const float* __restrict__ query,
                      float* __restrict__ out,
                      int nAtoms, int nQuery)
{
    __shared__ float4 atoms[MAX_ATOMS];   // {x, y, z, -0.5*L2E*|c|^2}

    const int tid    = threadIdx.x;
    const int lane   = tid & 31;
    const int wave   = tid >> 5;
    const int lane16 = lane & 15;
    const int hi     = lane >> 4;         // which 16-lane half

    // Stage atoms once per block; pad with -inf bias -> exp2(-inf) = 0 weight
    // (pad column is {0,0,0,-inf}: products are finite*0 and 1*(-inf) -> -inf, no NaN).
    if (tid < MAX_ATOMS) {
        float4 a4;
        if (tid < nAtoms) {
            float x = coords[3 * tid + 0];
            float y = coords[3 * tid + 1];
            float z = coords[3 * tid + 2];
            a4 = make_float4(x, y, z, -0.5f * L2E * (x * x + y * y + z * z));
        } else {
            a4 = make_float4(0.0f, 0.0f, 0.0f, -INFINITY);
        }
        atoms[tid] = a4;
    }
    __syncthreads();

    const int nTiles = (nQuery + 15) >> 4;
    const int tile   = blockIdx.x * WAVES_PER_BLOCK + wave;
    if (tile >= nTiles) return;           // wave-uniform: EXEC stays all-ones
    const int t16 = tile << 4;

    // This lane's query row m = lane16 (clamp only matters if M % 16 != 0).
    int qi = t16 + lane16;
    if (qi >= nQuery) qi = nQuery - 1;
    const float qx   = query[3 * qi + 0];
    const float qy   = query[3 * qi + 1];
    const float qz   = query[3 * qi + 2];
    const float hqn2 = -0.5f * L2E * (qx * qx + qy * qy + qz * qz);

    // A operand (16x4 f32): K = {L2E*x, L2E*y, L2E*z, 1}; VGPR0 = K0|K2, VGPR1 = K1|K3.
    v2f A2;
    A2.x = L2E * (hi ? qz : qx);
    A2.y = hi ? 1.0f : (L2E * qy);

    // Loop-invariant C operand: hq2v[r] = -0.5*L2E*|q_m|^2 for row m = r + 8*hi.
    v8f hq2v;
#pragma unroll
    for (int r = 0; r < 8; ++r)
        hq2v[r] = __shfl(hqn2, r + (hi << 3), 32);

    float ax[8], ay[8], az[8], aw[8];
#pragma unroll
    for (int r = 0; r < 8; ++r) { ax[r] = 0.f; ay[r] = 0.f; az[r] = 0.f; aw[r] = 0.f; }

    const int ng = (nAtoms + 15) >> 4;
    if (ng == 8) {                        // N = 128 fast path: 8 unrolled WMMAs
#pragma unroll
        for (int g = 0; g < 8; ++g) {
            float4 c4 = atoms[(g << 4) + lane16];   // ds_load_b128
            gnf_group(c4, A2, hi, hq2v, ax, ay, az, aw);
        }
    } else {
        for (int g = 0; g < ng; ++g) {
            float4 c4 = atoms[(g << 4) + lane16];
            gnf_group(c4, A2, hi, hq2v, ax, ay, az, aw);
        }
    }

    // Butterfly-reduce each row across the 16 lanes of its half (xor <= 8 stays in-half).
#pragma unroll
    for (int r = 0; r < 8; ++r) {
        float x = ax[r], y = ay[r], z = az[r], w = aw[r];
#pragma unroll
        for (int off = 8; off > 0; off >>= 1) {
            x += __shfl_xor(x, off, 32);
            y += __shfl_xor(y, off, 32);
            z += __shfl_xor(z, off, 32);
            w += __shfl_xor(w, off, 32);
        }
        const int m = r + (hi << 3);      // row held by this half's register r
        if (lane16 == m) {                // writer lane's own (qx,qy,qz) == q[t16+m]
            const int gm = t16 + m;
            if (gm < nQuery) {
                out[3 * gm + 0] = x - qx * w;   // sum(w*c) - q*sum(w)
                out[3 * gm + 1] = y - qy * w;
                out[3 * gm + 2] = z - qz * w;
            }
        }
    }
}

extern "C" void kernel_launch(void* const* d_in, const int* in_sizes, int n_in,
                              void* d_out, int out_size, void* d_ws, size_t ws_size,
                              hipStream_t stream) {
    const float* coords = (const float*)d_in[0];   // (N,3) f32
    const float* query  = (const float*)d_in[1];   // (M,3) f32
    float* out = (float*)d_out;                    // (M,3) f32

    const int nAtoms = in_sizes[0] / 3;
    const int nQuery = in_sizes[1] / 3;
    const int nTiles = (nQuery + 15) / 16;
    const int blocks = (nTiles + WAVES_PER_BLOCK - 1) / WAVES_PER_BLOCK;

    gnf_field_kernel<<<blocks, 256, 0, stream>>>(coords, query, out, nAtoms, nQuery);
}